// JointSampling_558345749189
// MI455X (gfx1250) — compile-verified
//
#include <hip/hip_runtime.h>

// ---- problem constants (from the reference) ----
constexpr int N_CATEG  = 32;
constexpr int CATEG_N  = 64;
constexpr int CONT_LEN = 512;
constexpr int DISC_LEN = N_CATEG * CATEG_N;        // 2048
constexpr int ROW_IN   = DISC_LEN + 2 * CONT_LEN;  // 3072 floats in per row
constexpr int ROW_SAMP = DISC_LEN + CONT_LEN;      // 2560 floats of `sample` per row
#define TAU_INV 2.0f                               // 1 / 0.5

// ---- counter-based RNG (deterministic per element) ----
__device__ __forceinline__ unsigned hash_u32(unsigned v) {
    v = v * 747796405u + 2891336453u;
    unsigned w = ((v >> ((v >> 28u) + 4u)) ^ v) * 277803737u;
    return (w >> 22u) ^ w;
}
// uniform in (0,1], never exactly 0; the largest h>>8 rounds up to exactly 1.0f
__device__ __forceinline__ float u01(unsigned h) {
    return (float)(h >> 8) * (1.0f / 16777216.0f) + (0.5f / 16777216.0f);
}

// ---- wave32 reductions ----
__device__ __forceinline__ float wave_max(float v) {
#pragma unroll
    for (int off = 16; off; off >>= 1) v = fmaxf(v, __shfl_xor(v, off, 32));
    return v;
}
__device__ __forceinline__ float wave_sum(float v) {
#pragma unroll
    for (int off = 16; off; off >>= 1) v += __shfl_xor(v, off, 32);
    return v;
}

__global__ __launch_bounds__(256)
void joint_sampling_kernel(const float* __restrict__ latent,
                           float* __restrict__ out,
                           int nrows) {
    __shared__ __align__(16) float smem[ROW_IN];   // 12 KB per workgroup

    const int row  = blockIdx.x;
    const int tid  = threadIdx.x;
    const int lane = tid & 31;
    const int wave = tid >> 5;

    const float* rowp = latent + (size_t)row * ROW_IN;

    // ---- stage the whole row into LDS with CDNA5 async global->LDS copies ----
    // 768 float4 = 12 KB; 256 threads x 3 b128 transfers, fully coalesced.
#pragma unroll
    for (int i = 0; i < 3; ++i) {
        const int idx4 = i * 256 + tid;  // float4 index within the row
        const unsigned lds_off = (unsigned)(uintptr_t)(&smem[0]) + (unsigned)(idx4 * 16);
        const float* gp = rowp + idx4 * 4;
        asm volatile("global_load_async_to_lds_b128 %0, %1, off"
                     :: "v"(lds_off), "v"(gp)
                     : "memory");
    }
    asm volatile("s_wait_asynccnt 0" ::: "memory");
    __syncthreads();

    // ================= discrete branch =================
    // 8 waves x 4 categories; each category = 64 floats, 2 per lane.
#pragma unroll
    for (int cc = 0; cc < N_CATEG / 8; ++cc) {
        const int c = wave * (N_CATEG / 8) + cc;
        const float2 a = *(const float2*)&smem[c * CATEG_N + lane * 2];

        // softmax #1 over the 64-wide category axis
        const float m  = wave_max(fmaxf(a.x, a.y));
        const float e0 = __expf(a.x - m);
        const float e1 = __expf(a.y - m);
        const float s  = wave_sum(e0 + e1);
        const float inv = __builtin_amdgcn_rcpf(s);
        const float p0 = e0 * inv, p1 = e1 * inv;

        // gumbel noise, then softmax #2 of (probs + g) / tau
        const unsigned base = (unsigned)row * (unsigned)DISC_LEN
                            + (unsigned)(c * CATEG_N + lane * 2);
        const float g0 = -__logf(-__logf(u01(hash_u32(base))));
        const float g1 = -__logf(-__logf(u01(hash_u32(base + 1u))));

        const float t0 = (p0 + g0) * TAU_INV;
        const float t1 = (p1 + g1) * TAU_INV;
        const float m2 = wave_max(fmaxf(t0, t1));
        float q0 = __expf(t0 - m2);
        float q1 = __expf(t1 - m2);
        const float s2   = wave_sum(q0 + q1);
        const float inv2 = __builtin_amdgcn_rcpf(s2);

        float2 r; r.x = q0 * inv2; r.y = q1 * inv2;
        *(float2*)&out[(size_t)row * ROW_SAMP + c * CATEG_N + lane * 2] = r;
    }

    // ================= continuous branch =================
    // Each thread handles 2 (mean, logvar) pairs: one float4 from LDS.
    const float4 mv = *(const float4*)&smem[DISC_LEN + tid * 4];
    const int i0 = tid * 2;

    const unsigned cbase = 0x40000000u
                         + (unsigned)row * (unsigned)(2 * CONT_LEN)
                         + (unsigned)(i0 * 2);
    const float twopi = 6.28318530718f;
    const float eps0 = __fsqrt_rn(-2.0f * __logf(u01(hash_u32(cbase + 0u))))
                     * __cosf(twopi * u01(hash_u32(cbase + 1u)));
    const float eps1 = __fsqrt_rn(-2.0f * __logf(u01(hash_u32(cbase + 2u))))
                     * __cosf(twopi * u01(hash_u32(cbase + 3u)));

    const float samp0 = mv.x + __expf(0.5f * mv.y) * eps0;
    const float samp1 = mv.z + __expf(0.5f * mv.w) * eps1;

    // sample[:, 2048 + i]
    *(float2*)&out[(size_t)row * ROW_SAMP + DISC_LEN + i0] = make_float2(samp0, samp1);
    // mean (starts at nrows*2560)
    *(float2*)&out[(size_t)nrows * ROW_SAMP + (size_t)row * CONT_LEN + i0] =
        make_float2(mv.x, mv.z);
    // logvar (starts at nrows*(2560+512))
    *(float2*)&out[(size_t)nrows * (ROW_SAMP + CONT_LEN) + (size_t)row * CONT_LEN + i0] =
        make_float2(mv.y, mv.w);
}

extern "C" void kernel_launch(void* const* d_in, const int* in_sizes, int n_in,
                              void* d_out, int out_size, void* d_ws, size_t ws_size,
                              hipStream_t stream) {
    const float* latent = (const float*)d_in[0];
    float* out = (float*)d_out;
    const int nrows = in_sizes[0] / ROW_IN;   // 16384 for the reference shapes
    hipLaunchKernelGGL(joint_sampling_kernel, dim3(nrows), dim3(256), 0, stream,
                       latent, out, nrows);
}